// GPModel_6107443495106
// MI455X (gfx1250) — compile-verified
//
#include <hip/hip_runtime.h>
#include <stdint.h>

// ---------------------------------------------------------------------------
// GP trajectory rollout for MI455X (gfx1250, wave32).
//  - fp32 WMMA (v_wmma_f32_16x16x4_f32) for Gram + Cholesky SYRK updates
//  - async global->LDS staging (global_load_async_to_lds_b128 / s_wait_asynccnt)
//  - 64x64 macro-tile SYRK: 4 waves share LDS panels, A-frag reused over 4 WMMAs
//  - blocked triangular solves with diag blocks in LDS for the rollout scan
// Workspace requirement: ~286 MB (G + 16 Cholesky factors + vectors).
// ---------------------------------------------------------------------------

#define N_TRAIN 2048
#define D_OUT   16
#define D_ACT   8
#define D_IN    24
#define HORIZON 64
#define NB      64          // Cholesky panel width
#define SPAD    68          // LDS row pitch (68*4B = 17 banks) -> conflict-free frag loads
#define JITTER  1e-6f

typedef __attribute__((ext_vector_type(2))) float v2f;
typedef __attribute__((ext_vector_type(8))) float v8f;

__device__ __forceinline__ v8f wmma4(v2f a, v2f b, v8f c) {
  // D = A(16x4,f32) * B(4x16,f32) + C(16x16,f32)
  return __builtin_amdgcn_wmma_f32_16x16x4_f32(false, a, false, b, (short)0, c,
                                               false, false);
}

// Async copy 16B global -> LDS (per-lane), tracked by ASYNCcnt.
__device__ __forceinline__ void async_copy_b128(const float* gsrc, const float* lds_dst) {
  uint32_t lofs = (uint32_t)(uintptr_t)lds_dst;  // flat LDS addr: low 32 bits = LDS offset
  asm volatile("global_load_async_to_lds_b128 %0, %1, off"
               :: "v"(lofs), "v"(gsrc) : "memory");
}
__device__ __forceinline__ void wait_async0() {
  asm volatile("s_wait_asynccnt 0" ::: "memory");
}

// ---------------------------------------------------------------------------
// Stage 0: row norms of x_train
__global__ void k_rownorm(const float* __restrict__ X, float* __restrict__ x2) {
  int i = blockIdx.x * blockDim.x + threadIdx.x;
  if (i < N_TRAIN) {
    float s = 0.f;
#pragma unroll
    for (int k = 0; k < D_IN; k++) { float v = X[i * D_IN + k]; s += v * v; }
    x2[i] = s;
  }
}

// Stage 1: G = X * X^T via f32 WMMA, K=24 = 6 steps of K=4. One wave per 16x16 tile.
__global__ void k_gram(const float* __restrict__ X, float* __restrict__ G) {
  int i0 = blockIdx.y * 16, j0 = blockIdx.x * 16;
  int l  = threadIdx.x;          // 0..31
  int m  = l & 15;
  int kh = l >> 4;               // 0 -> K+0/K+1 , 1 -> K+2/K+3  (ISA A-frag layout)
  v8f acc = {};
#pragma unroll
  for (int kk = 0; kk < D_IN; kk += 4) {
    v2f a, b;
    a.x = X[(i0 + m) * D_IN + kk + 2 * kh + 0];
    a.y = X[(i0 + m) * D_IN + kk + 2 * kh + 1];
    b.x = X[(j0 + m) * D_IN + kk + 2 * kh + 0];   // B = X(j-block)^T
    b.y = X[(j0 + m) * D_IN + kk + 2 * kh + 1];
    acc = wmma4(a, b, acc);
  }
#pragma unroll
  for (int r = 0; r < 8; r++) {                   // C/D layout: M = r + 8*kh, N = m
    G[(size_t)(i0 + r + (kh ? 8 : 0)) * N_TRAIN + (j0 + m)] = acc[r];
  }
}

// Stage 2: Ky_d = var_d * exp(-0.5*sq/ls_d^2) + (noise_d^2+jitter)*I  (16 matrices)
__global__ void k_assemble(const float* __restrict__ G, const float* __restrict__ x2,
                           const float* __restrict__ ls, const float* __restrict__ var,
                           const float* __restrict__ noise, float* __restrict__ L) {
  int j = blockIdx.x * blockDim.x + threadIdx.x;
  int i = blockIdx.y;
  int d = blockIdx.z;
  float ls2 = ls[d] * ls[d];
  float sq  = fmaxf(x2[i] + x2[j] - 2.f * G[(size_t)i * N_TRAIN + j], 0.f);
  float v   = var[d] * __expf(-0.5f * sq / ls2);
  if (i == j) v += noise[d] * noise[d] + JITTER;
  L[(size_t)d * N_TRAIN * N_TRAIN + (size_t)i * N_TRAIN + j] = v;
}

// ---------------------------------------------------------------------------
// Blocked right-looking Cholesky (batched over d = grid.x / grid.y / grid.z in potrf / trsm / syrk)

// 3a: factor 64x64 diagonal block in LDS (one block per d, 64 threads)
__global__ void k_potrf(float* __restrict__ L, int kb) {
  __shared__ float D[NB][NB];
  int d = blockIdx.x;
  float* A = L + (size_t)d * N_TRAIN * N_TRAIN;
  int r0 = kb * NB;
  int t  = threadIdx.x;
  for (int r = 0; r < NB; r++) D[r][t] = A[(size_t)(r0 + r) * N_TRAIN + r0 + t];
  __syncthreads();
  for (int j = 0; j < NB; j++) {
    if (t == j) D[j][j] = sqrtf(fmaxf(D[j][j], 1e-20f));
    __syncthreads();
    if (t > j) D[t][j] /= D[j][j];
    __syncthreads();
    if (t > j) {
      for (int c = j + 1; c <= t; c++) D[t][c] -= D[t][j] * D[c][j];
    }
    __syncthreads();
  }
  for (int r = 0; r < NB; r++)
    A[(size_t)(r0 + r) * N_TRAIN + r0 + t] = (t <= r) ? D[r][t] : 0.f;
}

// 3b: panel solve  L21 = A21 * L11^{-T}  (one row per thread, diag block in LDS)
__global__ void k_trsm(float* __restrict__ L, int kb) {
  __shared__ float D[NB][NB];
  int d  = blockIdx.y;
  float* A = L + (size_t)d * N_TRAIN * N_TRAIN;
  int r0 = kb * NB;
  int row = r0 + NB + blockIdx.x * blockDim.x + threadIdx.x;
  for (int idx = threadIdx.x; idx < NB * NB; idx += blockDim.x)
    D[idx >> 6][idx & 63] = A[(size_t)(r0 + (idx >> 6)) * N_TRAIN + r0 + (idx & 63)];
  __syncthreads();
  if (row < N_TRAIN) {
    float x[NB];
    float* Arow = &A[(size_t)row * N_TRAIN + r0];
#pragma unroll
    for (int j = 0; j < NB; j++) {
      float s = Arow[j];
#pragma unroll
      for (int p = 0; p < j; p++) s -= x[p] * D[j][p];
      x[j] = s / D[j][j];
    }
#pragma unroll
    for (int j = 0; j < NB; j++) Arow[j] = x[j];
  }
}

// 3c: trailing update  A22 -= L21 * L21^T  via f32 WMMA.
//     64x64 macro-tile per 128-thread block (4 waves). Panels staged to LDS once
//     via async b128 copies; each wave computes a 16x64 strip: one A-frag feeds
//     4 WMMAs (4 accumulators).  Lower-triangle macro-tiles only.
__global__ void k_syrk(float* __restrict__ L, int kb) {
  int r0 = kb * NB;
  int trail0 = r0 + NB;
  int bi = blockIdx.x, bj = blockIdx.y;
  if (bi < bj) return;                      // uniform branch: EXEC stays all-ones
  int d = blockIdx.z;
  float* A = L + (size_t)d * N_TRAIN * N_TRAIN;
  __shared__ float Pa[NB][SPAD];
  __shared__ float Pb[NB][SPAD];
  int t = threadIdx.x;                      // 128 threads = 4 waves
  int ra0 = trail0 + bi * NB;
  int rb0 = trail0 + bj * NB;
  // async stage both 64x64 panels; 16B per lane per issue, rows 16B-aligned
  {
    int lr = t >> 4;                        // 0..7  (row within group of 8)
    int lc = (t & 15) * 4;                  // 0..60
#pragma unroll
    for (int rr = 0; rr < NB; rr += 8) {
      async_copy_b128(&A[(size_t)(ra0 + rr + lr) * N_TRAIN + r0 + lc], &Pa[rr + lr][lc]);
      async_copy_b128(&A[(size_t)(rb0 + rr + lr) * N_TRAIN + r0 + lc], &Pb[rr + lr][lc]);
    }
  }
  int w  = t >> 5;                          // wave id -> 16-row strip
  int l  = t & 31;
  int m  = l & 15;
  int kh = l >> 4;
  // prefetch the C strip while the async engine fills LDS
#pragma unroll
  for (int n = 0; n < 4; n++)
    __builtin_prefetch(&A[(size_t)(ra0 + w * 16 + (kh ? 8 : 0)) * N_TRAIN + rb0 + n * 16 + m], 0, 3);
  wait_async0();
  __syncthreads();                          // panels visible to all 4 waves

  v8f acc0 = {}, acc1 = {}, acc2 = {}, acc3 = {};
#pragma unroll
  for (int kk = 0; kk < NB; kk += 4) {
    v2f a, b;
    a.x = Pa[w * 16 + m][kk + 2 * kh + 0];
    a.y = Pa[w * 16 + m][kk + 2 * kh + 1];
    b.x = Pb[ 0 + m][kk + 2 * kh + 0]; b.y = Pb[ 0 + m][kk + 2 * kh + 1];
    acc0 = wmma4(a, b, acc0);
    b.x = Pb[16 + m][kk + 2 * kh + 0]; b.y = Pb[16 + m][kk + 2 * kh + 1];
    acc1 = wmma4(a, b, acc1);
    b.x = Pb[32 + m][kk + 2 * kh + 0]; b.y = Pb[32 + m][kk + 2 * kh + 1];
    acc2 = wmma4(a, b, acc2);
    b.x = Pb[48 + m][kk + 2 * kh + 0]; b.y = Pb[48 + m][kk + 2 * kh + 1];
    acc3 = wmma4(a, b, acc3);
  }
#pragma unroll
  for (int r = 0; r < 8; r++) {
    int row = ra0 + w * 16 + r + (kh ? 8 : 0);
    size_t base = (size_t)row * N_TRAIN + rb0 + m;
    A[base +  0] -= acc0[r];
    A[base + 16] -= acc1[r];
    A[base + 32] -= acc2[r];
    A[base + 48] -= acc3[r];
  }
}

// ---------------------------------------------------------------------------
// Stage 4: alpha_d = Ky_d^{-1} y[:,d] via blocked fwd+bwd substitution.
__global__ void k_alpha(const float* __restrict__ L, const float* __restrict__ y,
                        float* __restrict__ Z, float* __restrict__ alpha) {
  int d = blockIdx.x;
  const float* A = L + (size_t)d * N_TRAIN * N_TRAIN;
  float* z  = Z + d * N_TRAIN;
  float* al = alpha + d * N_TRAIN;
  __shared__ float Dg[NB][NB];
  __shared__ float rhs[NB];
  __shared__ float red[256];
  int t = threadIdx.x;                      // 256 threads: 4 per row
  int rloc = t >> 2, lane4 = t & 3;
  // forward:  L z = y[:,d]
  for (int b = 0; b < N_TRAIN / NB; b++) {
    int r0 = b * NB;
    float s = 0.f;
    const float* Arow = &A[(size_t)(r0 + rloc) * N_TRAIN];
    for (int c = lane4; c < r0; c += 4) s += Arow[c] * z[c];
    red[t] = s;
    __syncthreads();
    if (lane4 == 0)
      rhs[rloc] = y[(r0 + rloc) * D_OUT + d] - (red[t] + red[t+1] + red[t+2] + red[t+3]);
    for (int idx = t; idx < NB * NB; idx += 256)
      Dg[idx >> 6][idx & 63] = A[(size_t)(r0 + (idx >> 6)) * N_TRAIN + r0 + (idx & 63)];
    __syncthreads();
    for (int j = 0; j < NB; j++) {
      if (t == 0) rhs[j] /= Dg[j][j];
      __syncthreads();
      if (t > j && t < NB) rhs[t] -= Dg[t][j] * rhs[j];
      __syncthreads();
    }
    if (t < NB) z[r0 + t] = rhs[t];
    __syncthreads();
  }
  // backward:  L^T alpha = z
  for (int b = N_TRAIN / NB - 1; b >= 0; b--) {
    int r0 = b * NB;
    float s = 0.f;
    for (int rr = r0 + NB + lane4; rr < N_TRAIN; rr += 4)
      s += A[(size_t)rr * N_TRAIN + r0 + rloc] * al[rr];
    red[t] = s;
    __syncthreads();
    if (lane4 == 0)
      rhs[rloc] = z[r0 + rloc] - (red[t] + red[t+1] + red[t+2] + red[t+3]);
    for (int idx = t; idx < NB * NB; idx += 256)
      Dg[idx >> 6][idx & 63] = A[(size_t)(r0 + (idx >> 6)) * N_TRAIN + r0 + (idx & 63)];
    __syncthreads();
    for (int j = NB - 1; j >= 0; j--) {
      if (t == 0) rhs[j] /= Dg[j][j];
      __syncthreads();
      if (t < j) rhs[t] -= Dg[j][t] * rhs[j];
      __syncthreads();
    }
    if (t < NB) al[r0 + t] = rhs[t];
    __syncthreads();
  }
}

__global__ void k_init(const float* __restrict__ s0, float* __restrict__ state) {
  int t = threadIdx.x;
  if (t < D_OUT) state[t] = s0[t];
}

// ---------------------------------------------------------------------------
// Stage 5 (per rollout step): one block per output d.
// kxs + mean, then blocked forward solve v = L^{-1} kxs with v resident in LDS,
// accumulate sum(v*v).  L triangles (134 MB total) live in the 192 MB L2.
__global__ void k_step_solve(const float* __restrict__ L, const float* __restrict__ Xtr,
                             const float* __restrict__ ls, const float* __restrict__ var,
                             const float* __restrict__ alpha,
                             const float* __restrict__ state,
                             const float* __restrict__ actions, int t_step,
                             float* __restrict__ mean, float* __restrict__ sumv2) {
  int d = blockIdx.x;
  const float* A = L + (size_t)d * N_TRAIN * N_TRAIN;
  __shared__ float kxs[N_TRAIN];            // rhs, solved in place -> v
  __shared__ float x[D_IN];
  __shared__ float Dg[NB][NB];
  __shared__ float red[512];
  __shared__ float rhs[NB];
  int t = threadIdx.x;                      // 512 threads
  if (t < D_OUT)      x[t] = state[t];
  else if (t < D_IN)  x[t] = actions[t_step * D_ACT + (t - D_OUT)];
  __syncthreads();
  float ls2 = ls[d] * ls[d], vr = var[d];
  float msum = 0.f;
  for (int i = t; i < N_TRAIN; i += 512) {
    float d2 = 0.f;
#pragma unroll
    for (int k = 0; k < D_IN; k++) { float df = Xtr[i * D_IN + k] - x[k]; d2 += df * df; }
    float kv = vr * __expf(-0.5f * d2 / ls2);
    kxs[i] = kv;
    msum  += kv * alpha[d * N_TRAIN + i];
  }
  red[t] = msum;
  __syncthreads();
  for (int s = 256; s > 0; s >>= 1) { if (t < s) red[t] += red[t + s]; __syncthreads(); }
  if (t == 0) mean[d] = red[0];
  __syncthreads();
  // blocked forward substitution, 8 threads per row for the GEMV update
  int rloc = t >> 3, l8 = t & 7;
  for (int b = 0; b < N_TRAIN / NB; b++) {
    int r0 = b * NB;
    float s = 0.f;
    const float* Arow = &A[(size_t)(r0 + rloc) * N_TRAIN];
    for (int c = l8; c < r0; c += 8) s += Arow[c] * kxs[c];
    red[t] = s;
    __syncthreads();
    if (l8 == 0) {
      float tot = 0.f;
      for (int q = 0; q < 8; q++) tot += red[(rloc << 3) + q];
      rhs[rloc] = kxs[r0 + rloc] - tot;
    }
    for (int idx = t; idx < NB * NB; idx += 512)
      Dg[idx >> 6][idx & 63] = A[(size_t)(r0 + (idx >> 6)) * N_TRAIN + r0 + (idx & 63)];
    __syncthreads();
    for (int j = 0; j < NB; j++) {
      if (t == 0) rhs[j] /= Dg[j][j];
      __syncthreads();
      if (t > j && t < NB) rhs[t] -= Dg[t][j] * rhs[j];
      __syncthreads();
    }
    if (t < NB) kxs[r0 + t] = rhs[t];
    __syncthreads();
  }
  float s2 = 0.f;
  for (int i = t; i < N_TRAIN; i += 512) s2 += kxs[i] * kxs[i];
  red[t] = s2;
  __syncthreads();
  for (int s = 256; s > 0; s >>= 1) { if (t < s) red[t] += red[t + s]; __syncthreads(); }
  if (t == 0) sumv2[d] = red[0];
}

// Stage 6: state update + outputs (trajectory | reward | cost | means | stds)
__global__ void k_finalize(const float* __restrict__ mean, const float* __restrict__ sumv2,
                           const float* __restrict__ var, const float* __restrict__ noise,
                           const float* __restrict__ eps, int t_step,
                           float* __restrict__ state, float* __restrict__ out) {
  __shared__ float ns[D_OUT];
  int t = threadIdx.x;
  if (t < D_OUT) {
    float vv  = var[t] - sumv2[t] + noise[t] * noise[t];
    float std = sqrtf(fmaxf(vv, 1e-12f));
    float m   = mean[t];
    float s   = m + std * eps[t_step * D_OUT + t];
    ns[t] = s;
    state[t] = s;
    out[t_step * D_OUT + t] = s;                                            // trajectory
    out[HORIZON * D_OUT + 2 * HORIZON + t_step * D_OUT + t] = m;            // means
    out[HORIZON * D_OUT + 2 * HORIZON + HORIZON * D_OUT + t_step * D_OUT + t] = std; // stds
  }
  __syncthreads();
  if (t == 0) {
    float r = 0.f, c = 0.f;
    for (int k = 0; k < D_OUT; k++) { r += ns[k]; c += ns[k] * ns[k]; }
    out[HORIZON * D_OUT + t_step] = r;                                      // reward
    out[HORIZON * D_OUT + HORIZON + t_step] = c;                            // cost
  }
}

// ---------------------------------------------------------------------------
extern "C" void kernel_launch(void* const* d_in, const int* in_sizes, int n_in,
                              void* d_out, int out_size, void* d_ws, size_t ws_size,
                              hipStream_t stream) {
  const float* x_train = (const float*)d_in[0];
  const float* y_train = (const float*)d_in[1];
  const float* ls      = (const float*)d_in[2];
  const float* var     = (const float*)d_in[3];
  const float* noise   = (const float*)d_in[4];
  const float* s0      = (const float*)d_in[5];
  const float* actions = (const float*)d_in[6];
  const float* eps     = (const float*)d_in[7];
  float* out = (float*)d_out;
  float* ws  = (float*)d_ws;

  const size_t NN = (size_t)N_TRAIN * N_TRAIN;
  float* G     = ws;                          // 16 MB
  float* L     = G + NN;                      // 256 MB (16 factors)
  float* X2    = L + (size_t)D_OUT * NN;
  float* ALPHA = X2 + N_TRAIN;
  float* Z     = ALPHA + D_OUT * N_TRAIN;
  float* STATE = Z + D_OUT * N_TRAIN;
  float* MEAN  = STATE + D_OUT;
  float* SUMV2 = MEAN + D_OUT;

  k_rownorm<<<(N_TRAIN + 255) / 256, 256, 0, stream>>>(x_train, X2);
  k_gram<<<dim3(N_TRAIN / 16, N_TRAIN / 16), 32, 0, stream>>>(x_train, G);
  k_assemble<<<dim3(N_TRAIN / 256, N_TRAIN, D_OUT), 256, 0, stream>>>(G, X2, ls, var, noise, L);

  for (int kb = 0; kb < N_TRAIN / NB; kb++) {
    k_potrf<<<D_OUT, NB, 0, stream>>>(L, kb);
    int trail = N_TRAIN - (kb + 1) * NB;
    if (trail > 0) {
      k_trsm<<<dim3((trail + 63) / 64, D_OUT), 64, 0, stream>>>(L, kb);
      int ntb = trail / NB;                 // trail is always a multiple of 64
      k_syrk<<<dim3(ntb, ntb, D_OUT), 128, 0, stream>>>(L, kb);
    }
  }

  k_alpha<<<D_OUT, 256, 0, stream>>>(L, y_train, Z, ALPHA);
  k_init<<<1, 32, 0, stream>>>(s0, STATE);

  for (int tt = 0; tt < HORIZON; tt++) {
    k_step_solve<<<D_OUT, 512, 0, stream>>>(L, x_train, ls, var, ALPHA, STATE,
                                            actions, tt, MEAN, SUMV2);
    k_finalize<<<1, 32, 0, stream>>>(MEAN, SUMV2, var, noise, eps, tt, STATE, out);
  }
}